// MaxPoolingAggregator_28767690949357
// MI455X (gfx1250) — compile-verified
//
#include <hip/hip_runtime.h>

// MI455X / gfx1250, wave32. WMMA f32 16x16x4.
typedef __attribute__((ext_vector_type(2))) float v2f;
typedef __attribute__((ext_vector_type(4))) float v4f;
typedef __attribute__((ext_vector_type(8))) float v8f;

#define WMMA_F32(a, b, c) \
  __builtin_amdgcn_wmma_f32_16x16x4_f32(false, (a), false, (b), (short)0, (c), false, false)

static constexpr int kB   = 20000;  // batch
static constexpr int kN   = 32;     // neighbors
static constexpr int kD   = 256;    // feature dim
static constexpr int kH   = 10;     // hidden (padded to 16)
static constexpr int kHP  = 16;     // padded hidden
static constexpr int kOut = 512;    // per-branch output dim

// -------------------------------------------------------------------------
// Kernel 1: h = relu(neigh @ W1 + b1); pooled[b][0..15] = max over 32 rows.
// One wave per b. A-frag trick: each lane owns (m = lane&15, c = lane>>4)
// and loads float4 of row m at cols 8i+4c .. 8i+4c+3; WMMA#1 of iter i uses
// elems {0,1}, WMMA#2 uses elems {2,3}. W1 is pre-swizzled into LDS with the
// *same* K-slot relabeling so the contraction pairs correct columns:
//   slot(v, half c) of iter i, wmma j  <->  actual k = 8i + 4c + 2j + v
// -------------------------------------------------------------------------
__global__ __launch_bounds__(256) void pool_kernel(
    const float* __restrict__ neigh, const float* __restrict__ W1,
    const float* __restrict__ b1, float* __restrict__ pooled) {
  __shared__ v2f fragW[64 * 32];  // 16 KB

  const int tid = threadIdx.x;
  for (int e = tid; e < 64 * 32; e += 256) {
    const int s = e >> 5, lane = e & 31;
    const int i = s >> 1, j = s & 1;
    const int c = lane >> 4, n = lane & 15;
    const int r = 8 * i + 4 * c + 2 * j;
    v2f w;
    w.x = (n < kH) ? W1[r * kH + n] : 0.0f;
    w.y = (n < kH) ? W1[(r + 1) * kH + n] : 0.0f;
    fragW[e] = w;
  }
  __syncthreads();

  const int lane = tid & 31;
  const int wave = tid >> 5;
  const int b    = blockIdx.x * 8 + wave;   // 2500 blocks * 8 waves = 20000
  const int m = lane & 15, c = lane >> 4;
  const int n = m;  // C-layout: lane&15 = output column
  const float bias = (n < kH) ? b1[n] : 0.0f;

  float mx = -3.0e38f;
#pragma unroll
  for (int tile = 0; tile < 2; ++tile) {
    v8f acc = {};
    const float* abase = neigh + ((size_t)b * kN + tile * 16 + m) * kD + 4 * c;
#pragma unroll
    for (int i = 0; i < 32; ++i) {
      // 655 MB read-once stream: non-temporal, don't pollute 192 MB L2.
      v4f a = __builtin_nontemporal_load((const v4f*)(abase + 8 * i));
      v2f b0  = fragW[(2 * i + 0) * 32 + lane];
      v2f b1v = fragW[(2 * i + 1) * 32 + lane];
      v2f a0 = {a.x, a.y}, a1 = {a.z, a.w};
      acc = WMMA_F32(a0, b0, acc);
      acc = WMMA_F32(a1, b1v, acc);
    }
#pragma unroll
    for (int r = 0; r < 8; ++r) mx = fmaxf(mx, acc[r] + bias);
  }
  mx = fmaxf(mx, __shfl_xor(mx, 16, 32));  // merge rows 0-7 / 8-15 halves
  mx = fmaxf(mx, 0.0f);                    // relu commutes with max
  if (lane < 16) pooled[(size_t)b * kHP + lane] = mx;
}

// -------------------------------------------------------------------------
// Kernel 2: out[:, 0:512] = self @ Wt2 + bt2.  (compute-bound, 5.24 GFLOP)
// All 10 waves of a block share one 16-column slice of Wt2, pre-swizzled
// into LDS in B-fragment layout (as kernel 1 does for W1). Inner loop per
// 2 WMMAs: one global_load_b128 (A) + conflict-free LDS fragment loads.
// grid = (32 ct, 125 mb); block = 320 (10 waves x 16 rows = 160 rows).
// -------------------------------------------------------------------------
__global__ __launch_bounds__(320) void out_self_kernel(
    const float* __restrict__ self, const float* __restrict__ Wt2,
    const float* __restrict__ bt2, float* __restrict__ out) {
  __shared__ v2f fragW[64 * 32];  // 16 KB: Wt2[:, ct*16 .. ct*16+15] swizzled

  const int tid = threadIdx.x;
  const int ct  = blockIdx.x;               // column tile 0..31
  for (int e = tid; e < 64 * 32; e += 320) {
    const int s = e >> 5, lane = e & 31;
    const int i = s >> 1, j = s & 1;
    const int c = lane >> 4, n = lane & 15;
    const int r = 8 * i + 4 * c + 2 * j;
    v2f w;
    w.x = Wt2[(size_t)r * kOut + ct * 16 + n];
    w.y = Wt2[(size_t)(r + 1) * kOut + ct * 16 + n];
    fragW[e] = w;
  }
  __syncthreads();

  const int lane = tid & 31;
  const int wave = tid >> 5;
  const int mbase = (blockIdx.y * 10 + wave) * 16;  // 125*160 = 20000 rows
  const int m = lane & 15, c = lane >> 4;
  const int n = m;
  const float bias = bt2[ct * 16 + n];

  v8f acc = {};
  const float* abase = self + ((size_t)(mbase + m)) * kD + 4 * c;
#pragma unroll
  for (int i = 0; i < 32; ++i) {
    v4f a = *(const v4f*)(abase + 8 * i);  // self is L2-resident (20 MB)
    v2f b0  = fragW[(2 * i + 0) * 32 + lane];
    v2f b1v = fragW[(2 * i + 1) * 32 + lane];
    v2f a0 = {a.x, a.y}, a1 = {a.z, a.w};
    acc = WMMA_F32(a0, b0, acc);
    acc = WMMA_F32(a1, b1v, acc);
  }

  const int outcol = ct * 16 + n;
  const int row0   = mbase + ((lane >= 16) ? 8 : 0);
#pragma unroll
  for (int r = 0; r < 8; ++r) {
    float v = acc[r] + bias;
    __builtin_nontemporal_store(v, &out[(size_t)(row0 + r) * 1024 + outcol]);
  }
}

// -------------------------------------------------------------------------
// Kernel 3: out[:, 512:1024] = pooled @ Wt1 + bt1.  (tiny: K=16 padded)
// One wave per 16x16 tile; 1250 M-tiles x 32 col-tiles = 40000 tiles.
// Wt1 rows >= 10 guarded to 0 (pooled pad cols are exact zeros too).
// -------------------------------------------------------------------------
__global__ __launch_bounds__(256) void out_neigh_kernel(
    const float* __restrict__ pooled, const float* __restrict__ Wt1,
    const float* __restrict__ bt1, float* __restrict__ out) {
  const int tid  = threadIdx.x;
  const int lane = tid & 31;
  const int wave = tid >> 5;
  const int t  = blockIdx.x * 8 + wave;  // 5000 blocks * 8 = 40000
  const int ct = t & 31;
  const int mt = t >> 5;                 // 0..1249
  const int m = lane & 15, c = lane >> 4;
  const int n = m;
  const int col = ct * 16 + n;
  const float bias = bt1[col];

  v8f acc = {};
  const float* abase = pooled + ((size_t)mt * 16 + m) * kHP + 4 * c;
#pragma unroll
  for (int i = 0; i < 2; ++i) {
    v4f a = *(const v4f*)(abase + 8 * i);
    const int r = 8 * i + 4 * c;
    v2f b0, b1v;
    b0.x  = (r + 0 < kH) ? Wt1[(size_t)(r + 0) * kOut + col] : 0.0f;
    b0.y  = (r + 1 < kH) ? Wt1[(size_t)(r + 1) * kOut + col] : 0.0f;
    b1v.x = (r + 2 < kH) ? Wt1[(size_t)(r + 2) * kOut + col] : 0.0f;
    b1v.y = (r + 3 < kH) ? Wt1[(size_t)(r + 3) * kOut + col] : 0.0f;
    v2f a0 = {a.x, a.y}, a1 = {a.z, a.w};
    acc = WMMA_F32(a0, b0, acc);
    acc = WMMA_F32(a1, b1v, acc);
  }

  const int outcol = 512 + col;
  const int row0   = mt * 16 + ((lane >= 16) ? 8 : 0);
#pragma unroll
  for (int r = 0; r < 8; ++r) {
    float v = acc[r] + bias;
    __builtin_nontemporal_store(v, &out[(size_t)(row0 + r) * 1024 + outcol]);
  }
}

extern "C" void kernel_launch(void* const* d_in, const int* in_sizes, int n_in,
                              void* d_out, int out_size, void* d_ws, size_t ws_size,
                              hipStream_t stream) {
  const float* self  = (const float*)d_in[0];  // [20000,256]
  const float* neigh = (const float*)d_in[1];  // [20000,32,256]
  const float* W1    = (const float*)d_in[2];  // [256,10]
  const float* b1    = (const float*)d_in[3];  // [10]
  const float* Wt1   = (const float*)d_in[4];  // [10,512]
  const float* bt1   = (const float*)d_in[5];  // [512]
  const float* Wt2   = (const float*)d_in[6];  // [256,512]
  const float* bt2   = (const float*)d_in[7];  // [512]
  float* out    = (float*)d_out;               // [20000,1024]
  float* pooled = (float*)d_ws;                // 20000*16*4 = 1.28 MB scratch

  pool_kernel<<<dim3(kB / 8), dim3(256), 0, stream>>>(neigh, W1, b1, pooled);
  out_self_kernel<<<dim3(kOut / 16, kB / 160), dim3(320), 0, stream>>>(
      self, Wt2, bt2, out);
  out_neigh_kernel<<<dim3((kB / 16) * (kOut / 16) / 8), dim3(256), 0, stream>>>(
      pooled, Wt1, bt1, out);
}